// SAGEEncoder_48661979464279
// MI455X (gfx1250) — compile-verified
//
#include <hip/hip_runtime.h>
#include <hip/hip_bf16.h>
#include <math.h>

typedef __attribute__((ext_vector_type(2))) float v2f;
typedef __attribute__((ext_vector_type(8))) float v8f;

#define EPS_BN 1e-5f

// ---------------- utility kernels ----------------

__global__ void zero_f32(float* __restrict__ p, long long n) {
    long long i = (long long)blockIdx.x * blockDim.x + threadIdx.x;
    if (i < n) p[i] = 0.0f;
}

// count in-degree (float, matching reference segment_sum of ones)
__global__ void degree_k(const int* __restrict__ ei, float* __restrict__ cnt, int nedges) {
    int e = blockIdx.x * blockDim.x + threadIdx.x;
    if (e < nedges) atomicAdd(&cnt[ei[nedges + e]], 1.0f);
}

// scatter-add features along edges: agg[dst] += X[src]; grid.y indexes float4 chunk
__global__ void scatter_add(const float* __restrict__ X, const int* __restrict__ ei,
                            float* __restrict__ agg, int nedges, int F) {
    int e = blockIdx.x * blockDim.x + threadIdx.x;
    if (e >= nedges) return;
    int c = blockIdx.y * 4;
    long long s = ei[e];
    long long d = ei[nedges + e];
    const float4 v = *(const float4*)(X + s * F + c);
    float* p = agg + d * F + c;
    atomicAdd(p + 0, v.x);
    atomicAdd(p + 1, v.y);
    atomicAdd(p + 2, v.z);
    atomicAdd(p + 3, v.w);
}

// agg[n][:] /= max(cnt[n], 1)
__global__ void div_cnt(float* __restrict__ agg, const float* __restrict__ cnt,
                        int nrows, int F) {
    int n = blockIdx.x * blockDim.x + threadIdx.x;
    if (n >= nrows) return;
    float inv = 1.0f / fmaxf(cnt[n], 1.0f);
    float* p = agg + (size_t)n * F;
    for (int f = 0; f < F; f += 4) {
        float4 v = *(float4*)(p + f);
        v.x *= inv; v.y *= inv; v.z *= inv; v.w *= inv;
        *(float4*)(p + f) = v;
    }
}

// pack Wl -> [K][32] zero-padded, (Wr+Ws) -> [K][32] zero-padded, bias = bl+bs padded to 32
__global__ void pack_weights(const float* __restrict__ Wl, const float* __restrict__ Wr,
                             const float* __restrict__ Ws, const float* __restrict__ bl,
                             const float* __restrict__ bs,
                             float* __restrict__ Wlp, float* __restrict__ Wrsp,
                             float* __restrict__ biasp, int K, int D) {
    int idx = blockIdx.x * blockDim.x + threadIdx.x;
    if (idx < 32) biasp[idx] = (idx < D) ? (bl[idx] + bs[idx]) : 0.0f;
    if (idx >= K * 32) return;
    int k = idx >> 5, c = idx & 31;
    float wl = 0.0f, wrs = 0.0f;
    if (c < D) {
        wl  = Wl[k * D + c];
        wrs = Wr[k * D + c] + Ws[k * D + c];
    }
    Wlp[idx]  = wl;
    Wrsp[idx] = wrs;
}

// ---------------- WMMA fused SAGE linear ----------------
// out[N x Dout] = agg @ Wl_pad + xin @ Wrs_pad + bias   (fp32 WMMA 16x16x4)
// One wave per 16-row tile; two 16-col accumulator tiles cover Dout<=32.
__global__ void __launch_bounds__(128)
sage_wmma(const float* __restrict__ agg, const float* __restrict__ xin,
          const float* __restrict__ Wlp, const float* __restrict__ Wrsp,
          const float* __restrict__ biasp, float* __restrict__ out,
          int nrows, int K, int Dout) {
    const int lane = threadIdx.x & 31;
    const int wave = threadIdx.x >> 5;
    const int tile = blockIdx.x * (blockDim.x >> 5) + wave;
    const int m0 = tile * 16;
    if (m0 >= nrows) return;               // wave-uniform: EXEC stays all-ones for WMMA

    const int half = lane >> 4;            // 0: lanes 0-15, 1: lanes 16-31
    const int l16 = lane & 15;
    const int kofs = half * 2;             // upper half-wave holds K+2,K+3

    v8f c0 = {};                           // cols 0..15
    v8f c1 = {};                           // cols 16..31 (zero-padded weights)
    const float* arow = agg + (size_t)(m0 + l16) * K;
    const float* xrow = xin + (size_t)(m0 + l16) * K;

    for (int k0 = 0; k0 < K; k0 += 4) {
        // A fragments (16x4): lane half selects K pair
        v2f a1, a2;
        a1.x = arow[k0 + kofs];     a1.y = arow[k0 + kofs + 1];
        a2.x = xrow[k0 + kofs];     a2.y = xrow[k0 + kofs + 1];
        // B fragments (4x16) from padded [K][32] weights
        const float* bl = Wlp  + (size_t)(k0 + kofs) * 32;
        const float* br = Wrsp + (size_t)(k0 + kofs) * 32;
        v2f bL0, bL1, bR0, bR1;
        bL0.x = bl[l16];      bL0.y = bl[32 + l16];
        bL1.x = bl[16 + l16]; bL1.y = bl[48 + l16];
        bR0.x = br[l16];      bR0.y = br[32 + l16];
        bR1.x = br[16 + l16]; bR1.y = br[48 + l16];

        c0 = __builtin_amdgcn_wmma_f32_16x16x4_f32(false, a1, false, bL0, (short)0, c0, false, false);
        c0 = __builtin_amdgcn_wmma_f32_16x16x4_f32(false, a2, false, bR0, (short)0, c0, false, false);
        c1 = __builtin_amdgcn_wmma_f32_16x16x4_f32(false, a1, false, bL1, (short)0, c1, false, false);
        c1 = __builtin_amdgcn_wmma_f32_16x16x4_f32(false, a2, false, bR1, (short)0, c1, false, false);
    }

    // C/D layout: vgpr i -> row m0+i (lanes 0-15) / m0+8+i (lanes 16-31); col = lane%16
    const int rbase = m0 + half * 8;
    const int col2 = 16 + l16;
#pragma unroll
    for (int i = 0; i < 8; ++i) {
        const int r = rbase + i;
        out[(size_t)r * Dout + l16] = c0[i] + biasp[l16];
        if (col2 < Dout) out[(size_t)r * Dout + col2] = c1[i] + biasp[col2];
    }
}

// ---------------- small dense linear: out = H @ W + b ----------------
__global__ void linear20(const float* __restrict__ H, const float* __restrict__ W,
                         const float* __restrict__ b, float* __restrict__ out,
                         int nrows, int Din, int Dout) {
    int n = blockIdx.x * blockDim.x + threadIdx.x;
    if (n >= nrows) return;
    float h[20];
    for (int k = 0; k < Din; ++k) h[k] = H[(size_t)n * Din + k];
    for (int j = 0; j < Dout; ++j) {
        float acc = b[j];
        for (int k = 0; k < Din; ++k) acc += h[k] * W[k * Dout + j];
        out[(size_t)n * Dout + j] = acc;
    }
}

// ---------------- batch-norm statistics (sum, sumsq per column) ----------------
__global__ void bn_stats(const float* __restrict__ H, float* __restrict__ stats,
                         int nrows, int D) {
    __shared__ float ls[64];
    float sum[20], sq[20];
    for (int j = 0; j < 20; ++j) { sum[j] = 0.0f; sq[j] = 0.0f; }
    for (int n = blockIdx.x * blockDim.x + threadIdx.x; n < nrows;
         n += gridDim.x * blockDim.x) {
        const float* p = H + (size_t)n * D;
        for (int j = 0; j < D; ++j) {
            float v = p[j];
            sum[j] += v;
            sq[j] += v * v;
        }
    }
    if (threadIdx.x < 64) ls[threadIdx.x] = 0.0f;
    __syncthreads();
    for (int j = 0; j < D; ++j) {
        atomicAdd(&ls[j], sum[j]);
        atomicAdd(&ls[32 + j], sq[j]);
    }
    __syncthreads();
    if (threadIdx.x < 64) atomicAdd(&stats[threadIdx.x], ls[threadIdx.x]);
}

// ---------------- batch-norm apply + ReLU ----------------
__global__ void bn_relu(const float* __restrict__ H, const float* __restrict__ stats,
                        const float* __restrict__ g, const float* __restrict__ be,
                        float* __restrict__ out, int nrows, int D) {
    int idx = blockIdx.x * blockDim.x + threadIdx.x;
    if (idx >= nrows * D) return;
    int j = idx % D;
    float invn = 1.0f / (float)nrows;
    float mu = stats[j] * invn;
    float var = stats[32 + j] * invn - mu * mu;   // biased variance
    float v = (H[idx] - mu) * rsqrtf(var + EPS_BN) * g[j] + be[j];
    out[idx] = v > 0.0f ? v : 0.0f;
}

// ---------------- host orchestration ----------------

extern "C" void kernel_launch(void* const* d_in, const int* in_sizes, int n_in,
                              void* d_out, int out_size, void* d_ws, size_t ws_size,
                              hipStream_t stream) {
    (void)in_sizes; (void)n_in; (void)out_size; (void)ws_size;
    constexpr int N = 100000, E = 1600000, F_IN = 64, D = 20, D_OUT = 16;

    const float* x   = (const float*)d_in[0];
    const int*   ei  = (const int*)d_in[1];   // [2][E]: src row then dst row
    const float* Wl1 = (const float*)d_in[3];
    const float* bl1 = (const float*)d_in[4];
    const float* Wr1 = (const float*)d_in[5];
    const float* Ws1 = (const float*)d_in[6];
    const float* bs1 = (const float*)d_in[7];
    const float* Wn1 = (const float*)d_in[8];
    const float* bn1 = (const float*)d_in[9];
    const float* g1  = (const float*)d_in[10];
    const float* be1 = (const float*)d_in[11];
    const float* Wlm = (const float*)d_in[12];
    const float* blm = (const float*)d_in[13];
    const float* Wrm = (const float*)d_in[14];
    const float* Wsm = (const float*)d_in[15];
    const float* bsm = (const float*)d_in[16];
    const float* Wnm = (const float*)d_in[17];
    const float* bnm = (const float*)d_in[18];
    const float* gm  = (const float*)d_in[19];
    const float* bem = (const float*)d_in[20];
    const float* Wl5 = (const float*)d_in[21];
    const float* bl5 = (const float*)d_in[22];
    const float* Wr5 = (const float*)d_in[23];
    const float* Ws5 = (const float*)d_in[24];
    const float* bs5 = (const float*)d_in[25];
    const float* Wn5 = (const float*)d_in[26];
    const float* bn5 = (const float*)d_in[27];

    // workspace layout (floats)
    float* ws    = (float*)d_ws;
    float* agg   = ws;                          // N*64 max
    float* cnt   = agg + (size_t)N * 64;        // N
    float* hA    = cnt + N;                     // N*20 (layer input h)
    float* hB    = hA + (size_t)N * D;          // N*20 (post-SAGE)
    float* hC    = hB + (size_t)N * D;          // N*20 (post-lin, pre-BN)
    float* stats = hC + (size_t)N * D;          // 64
    float* Wlp   = stats + 64;                  // 64*32 max
    float* Wrsp  = Wlp + 64 * 32;               // 64*32 max
    float* biasp = Wrsp + 64 * 32;              // 32

    const int TB = 256;
    const int wavesPerBlock = 4;                // 128-thread blocks for WMMA kernel

    // in-degree (shared by all layers)
    zero_f32<<<(N + TB - 1) / TB, TB, 0, stream>>>(cnt, N);
    degree_k<<<(E + TB - 1) / TB, TB, 0, stream>>>(ei, cnt, E);

    // fused SAGE step: agg=mean_agg(hin); hout = agg@Wl + hin@(Wr+Ws) + (bl+bs)
    auto sage_step = [&](const float* hin, int K,
                         const float* Wl, const float* Wr, const float* Ws,
                         const float* bl, const float* bs, float* hout) {
        zero_f32<<<((long long)N * K + TB - 1) / TB, TB, 0, stream>>>(agg, (long long)N * K);
        scatter_add<<<dim3((E + TB - 1) / TB, K / 4), TB, 0, stream>>>(hin, ei, agg, E, K);
        div_cnt<<<(N + TB - 1) / TB, TB, 0, stream>>>(agg, cnt, N, K);
        pack_weights<<<(K * 32 + TB - 1) / TB, TB, 0, stream>>>(Wl, Wr, Ws, bl, bs,
                                                                Wlp, Wrsp, biasp, K, D);
        int tiles = N / 16;                                         // N % 16 == 0
        int blocks = (tiles + wavesPerBlock - 1) / wavesPerBlock;
        sage_wmma<<<blocks, wavesPerBlock * 32, 0, stream>>>(agg, hin, Wlp, Wrsp, biasp,
                                                             hout, N, K, D);
    };

    auto bn_block = [&](const float* Wn, const float* bn, const float* g,
                        const float* be, float* hout) {
        linear20<<<(N + TB - 1) / TB, TB, 0, stream>>>(hB, Wn, bn, hC, N, D, D);
        zero_f32<<<1, 64, 0, stream>>>(stats, 64);
        bn_stats<<<512, TB, 0, stream>>>(hC, stats, N, D);
        bn_relu<<<(N * D + TB - 1) / TB, TB, 0, stream>>>(hC, stats, g, be, hout, N, D);
    };

    // layer 1 (K = F_IN = 64)
    sage_step(x, F_IN, Wl1, Wr1, Ws1, bl1, bs1, hB);
    bn_block(Wn1, bn1, g1, be1, hA);

    // layers 2-4 (K = D = 20)
    for (int i = 0; i < 3; ++i) {
        sage_step(hA, D, Wlm + i * D * D, Wrm + i * D * D, Wsm + i * D * D,
                  blm + i * D, bsm + i * D, hB);
        bn_block(Wnm + i * D * D, bnm + i * D, gm + i * D, bem + i * D, hA);
    }

    // layer 5: SAGE + skip, then final linear (D -> D_OUT), no BN/ReLU
    sage_step(hA, D, Wl5, Wr5, Ws5, bl5, bs5, hB);
    linear20<<<(N + TB - 1) / TB, TB, 0, stream>>>(hB, Wn5, bn5, (float*)d_out,
                                                   N, D, D_OUT);
}